// GroupMLP_31473520345758
// MI455X (gfx1250) — compile-verified
//
#include <hip/hip_runtime.h>
#include <float.h>

// Problem constants (match reference)
#define BB   8
#define NN   4096
#define CC   128
#define KK   16
#define EPSV 1e-5f

typedef __attribute__((ext_vector_type(2))) float v2f;
typedef __attribute__((ext_vector_type(8))) float v8f;

// ---------------------------------------------------------------------------
// Kernel 1: fused KNN (distance compute + top-16 selection), one batch's xyz
// resident in LDS as float4 (x,y,z,|p|^2). Same d2 formula as the reference
// (s_i + s_j - 2*dot) so tie ordering matches; insertion uses strict '<' so
// equal distances keep the earlier index (top_k tie behavior).
// ---------------------------------------------------------------------------
__global__ __launch_bounds__(256) void knn_kernel(const float* __restrict__ xyz,
                                                  int* __restrict__ idx) {
    __shared__ float4 pts[NN];                       // 64 KB LDS
    const int b = blockIdx.x >> 4;                   // 16 blocks per batch
    const int i = ((blockIdx.x & 15) << 8) + threadIdx.x;
    const float* base = xyz + (size_t)b * NN * 3;

    for (int p = threadIdx.x; p < NN; p += 256) {
        float px = base[p * 3 + 0];
        float py = base[p * 3 + 1];
        float pz = base[p * 3 + 2];
        pts[p] = make_float4(px, py, pz, px * px + py * py + pz * pz);
    }
    __syncthreads();

    const float4 me = pts[i];
    float d[KK];
    int   id[KK];
#pragma unroll
    for (int t = 0; t < KK; ++t) { d[t] = FLT_MAX; id[t] = 0; }

    for (int j = 0; j < NN; ++j) {
        float4 p  = pts[j];                          // LDS broadcast read
        float dot = me.x * p.x + me.y * p.y + me.z * p.z;
        float dj  = me.w + p.w - 2.0f * dot;
        if (dj < d[KK - 1]) {                        // beats current worst?
            float cur = dj; int cid = j;             // branchless ripple insert
#pragma unroll
            for (int t = 0; t < KK; ++t) {
                bool  sw  = cur < d[t];
                float od  = d[t];  int oid = id[t];
                d[t]  = sw ? cur : od;
                id[t] = sw ? cid : oid;
                cur   = sw ? od  : cur;
                cid   = sw ? oid : cid;
            }
        }
    }

    int* orow = idx + ((size_t)b * NN + i) * KK;
#pragma unroll
    for (int t = 0; t < KK; ++t) orow[t] = id[t];
}

// ---------------------------------------------------------------------------
// Kernel 2: a = relu(BN(W @ x)) per batch via V_WMMA_F32_16X16X4_F32.
// One wave per 16x16 output tile; 32 WMMA steps over C=128.
//   A (16x4 f32): lane m=lane&15 holds W[o_base+m, c0+2*half ..+1] (float2)
//   B (4x16 f32): lane holds x[c0+2*half, j], x[c0+2*half+1, j]
//   C/D (16x16 f32): VGPR v -> row v + 8*half, col lane&15
// ---------------------------------------------------------------------------
__global__ __launch_bounds__(256) void gemm_bn_relu_kernel(
    const float* __restrict__ x, const float* __restrict__ W,
    const float* __restrict__ gamma, const float* __restrict__ beta,
    const float* __restrict__ mean,  const float* __restrict__ var,
    float* __restrict__ abuf) {
    const int lane = threadIdx.x & 31;
    const int wave = threadIdx.x >> 5;
    const int tile = blockIdx.x * 8 + wave;          // B * (C/16) * (N/16) tiles
    const int jt   = tile & 255;                     // N/16 = 256
    const int ot   = (tile >> 8) & 7;                // C/16 = 8
    const int b    = tile >> 11;

    const int m    = lane & 15;
    const int half = lane >> 4;                      // 0: K=0,1   1: K=2,3

    const float* Wrow = W + (size_t)(ot * 16 + m) * CC;
    const float* xcol = x + (size_t)b * CC * NN + (size_t)jt * 16 + m;

    v8f acc = {};
#pragma unroll
    for (int c0 = 0; c0 < CC; c0 += 4) {
        const int k0 = c0 + 2 * half;
        v2f av = *(const v2f*)(Wrow + k0);           // contiguous, 8B aligned
        v2f bv;
        bv.x = xcol[(size_t)k0 * NN];
        bv.y = xcol[(size_t)(k0 + 1) * NN];
        acc = __builtin_amdgcn_wmma_f32_16x16x4_f32(
            /*neg_a=*/false, av, /*neg_b=*/false, bv,
            /*c_mod=*/(short)0, acc, /*reuse_a=*/false, /*reuse_b=*/false);
    }

#pragma unroll
    for (int v = 0; v < 8; ++v) {
        const int o = ot * 16 + v + 8 * half;
        float s  = gamma[o] * rsqrtf(var[o] + EPSV);
        float bi = beta[o] - mean[o] * s;
        float val = fmaxf(acc[v] * s + bi, 0.0f);
        abuf[((size_t)b * CC + o) * NN + (size_t)jt * 16 + m] = val;
    }
}

// ---------------------------------------------------------------------------
// Kernel 3: out[b,c,n] = max_k a[b,c,idx[b,n,k]]. a (16.7MB) is L2-resident.
// Thread id == flat [B,C,N] index -> coalesced output stores; idx row read
// as four int4 (rows are 64B aligned).
// ---------------------------------------------------------------------------
__global__ __launch_bounds__(256) void gather_max_kernel(
    const float* __restrict__ abuf, const int* __restrict__ idx,
    float* __restrict__ out) {
    const int g = blockIdx.x * 256 + threadIdx.x;    // over B*C*N
    const int n = g & (NN - 1);
    const int c = (g >> 12) & (CC - 1);
    const int b = g >> 19;

    const int4* row = (const int4*)(idx + ((size_t)b * NN + n) * KK);
    const float* ab = abuf + ((size_t)b * CC + c) * NN;

    float mx = -FLT_MAX;
#pragma unroll
    for (int q = 0; q < 4; ++q) {
        int4 r = row[q];
        mx = fmaxf(mx, ab[r.x]);
        mx = fmaxf(mx, ab[r.y]);
        mx = fmaxf(mx, ab[r.z]);
        mx = fmaxf(mx, ab[r.w]);
    }
    out[g] = mx;
}

// ---------------------------------------------------------------------------
extern "C" void kernel_launch(void* const* d_in, const int* in_sizes, int n_in,
                              void* d_out, int out_size, void* d_ws, size_t ws_size,
                              hipStream_t stream) {
    const float* xyz   = (const float*)d_in[0];
    const float* x     = (const float*)d_in[1];
    const float* W     = (const float*)d_in[2];
    const float* gamma = (const float*)d_in[3];
    const float* beta  = (const float*)d_in[4];
    const float* mean  = (const float*)d_in[5];
    const float* var   = (const float*)d_in[6];
    float* out = (float*)d_out;

    int*   idx  = (int*)d_ws;                                        // 2 MB
    float* abuf = (float*)((char*)d_ws + (size_t)BB * NN * KK * 4);  // 16.7 MB

    // 1) KNN: 16 blocks per batch, 256 rows each
    knn_kernel<<<BB * 16, 256, 0, stream>>>(xyz, idx);
    // 2) z = ReLU(BN(W@x)) : B*(C/16)*(N/16) = 16384 tiles, 8 waves/block
    gemm_bn_relu_kernel<<<16384 / 8, 256, 0, stream>>>(x, W, gamma, beta,
                                                       mean, var, abuf);
    // 3) gather-max over 16 neighbors
    gather_max_kernel<<<(BB * CC * NN) / 256, 256, 0, stream>>>(abuf, idx, out);
}